// PhysicsInformedNN_12086037970987
// MI455X (gfx1250) — compile-verified
//
#include <hip/hip_runtime.h>
#include <hip/hip_bf16.h>
#include <math.h>

#define NPTS 8192
#define HID 256
#define NHIDL 7
#define TPB 128                       // 4 waves of 32
#define WAVES 4
#define PTS_PER_BLK 8                 // 2 points per wave
#define NBLK (NPTS / PTS_PER_BLK)     // 1024
#define KSLAB 32
#define WSTRIDE 36                    // padded K stride for staged W (dwords)
#define ASTRIDE 260                   // padded U stride for activations (dwords)
#define ACT_DW (WAVES * 16 * ASTRIDE) // 16640 dwords
#define WL_DW  (HID * WSTRIDE)        // 9216 dwords per W buffer
#define SMEM_BYTES ((ACT_DW + 2 * WL_DW) * 4)   // 140288 B dynamic LDS

typedef float v2f __attribute__((ext_vector_type(2)));
typedef float v8f __attribute__((ext_vector_type(8)));

// ---------------------------------------------------------------- min/max ---
__global__ void __launch_bounds__(256) minmax_kernel(const float* __restrict__ X,
                                                     float* __restrict__ ws) {
    __shared__ float smn[256], smx[256];
    const int tid = threadIdx.x;
    float mn[4], mx[4];
#pragma unroll
    for (int k = 0; k < 4; ++k) { mn[k] = 3.4e38f; mx[k] = -3.4e38f; }
    for (int i = tid; i < NPTS; i += 256) {
#pragma unroll
        for (int k = 0; k < 4; ++k) {
            float v = X[i * 4 + k];
            mn[k] = fminf(mn[k], v);
            mx[k] = fmaxf(mx[k], v);
        }
    }
    for (int k = 0; k < 4; ++k) {
        smn[tid] = mn[k]; smx[tid] = mx[k];
        __syncthreads();
        for (int off = 128; off > 0; off >>= 1) {
            if (tid < off) {
                smn[tid] = fminf(smn[tid], smn[tid + off]);
                smx[tid] = fmaxf(smx[tid], smx[tid + off]);
            }
            __syncthreads();
        }
        if (tid == 0) { ws[k] = smn[0]; ws[4 + k] = smx[0]; }
        __syncthreads();
    }
}

// ------------------------------------------------------------ main kernel ---
__global__ void __launch_bounds__(TPB) pinn_kernel(
    const float* __restrict__ X, const float* __restrict__ uvw,
    const float* __restrict__ W_in, const float* __restrict__ b_in,
    const float* __restrict__ W_hid, const float* __restrict__ b_hid,
    const float* __restrict__ W_out, const float* __restrict__ b_out,
    const float* __restrict__ Re, const float* __restrict__ mm,
    float* __restrict__ partials)
{
    extern __shared__ float smem[];
    float* actL = smem;                 // [wave][16 rows][ASTRIDE]
    float* wL0  = smem + ACT_DW;        // W slab buffer 0 (also reused at the end)

    const int tid  = threadIdx.x;
    const int lane = tid & 31;
    const int wv   = tid >> 5;
    const int l15  = lane & 15;
    const int ph   = lane >> 4;         // which of this wave's two points
    const int kd   = ph * 2;            // WMMA K sub-offset for upper half-wave
    const int myp  = blockIdx.x * PTS_PER_BLK + wv * 2 + ph;

    // ---- async transpose-stage of one K-slab of W into LDS buffer `buf` ----
    // 64 x GLOBAL_LOAD_ASYNC_TO_LDS_B32 per wave: global (u,n) -> LDS [n][u].
    auto stage = [&](const float* Wl, int s, int buf) {
        const unsigned long long gbase =
            (unsigned long long)(uintptr_t)(Wl + s * KSLAB * HID);
        const unsigned int ldsLane = (unsigned int)((ACT_DW + buf * WL_DW) * 4 + lane * 144);
        const unsigned int gLane   = (unsigned int)(lane * 4);
#pragma unroll 8
        for (int i = 0; i < 64; ++i) {
            const int idx = wv * 64 + i;       // 256 chunks over 4 waves
            const int u  = idx >> 3;           // 0..31 (k within slab)
            const int nb = idx & 7;            // 0..7  (n block of 32)
            unsigned int va = gLane + (unsigned int)(u * 1024 + nb * 128);       // bytes
            unsigned int ld = ldsLane + (unsigned int)(nb * 4608 + u * 4);       // bytes
            asm volatile("global_load_async_to_lds_b32 %0, %1, %2"
                         :: "v"(ld), "v"(va), "s"(gbase) : "memory");
        }
    };

    // input normalization + tangent seeds
    float s4[4], xn[4];
#pragma unroll
    for (int k = 0; k < 4; ++k) {
        float lo = mm[k], hi = mm[4 + k];
        s4[k] = 2.0f / (hi - lo);
        xn[k] = (X[myp * 4 + k] - lo) * s4[k] - 1.0f;
    }

    // kick off the first W slab while we do the input layer
    stage(W_hid, 0, 0);

    // Accumulators: acc[t] covers units u = 16t + (lane&15); VGPR r = jet channel r.
    // ch0 = value, ch1..4 = d/dx_k, ch5..7 = d^2/dx_k^2 (k=0..2)
    v8f acc[16];

    // ---- input layer (4 -> 256, done per lane) ----
#pragma unroll
    for (int t = 0; t < 16; ++t) {
        const int u = t * 16 + l15;
        float w0 = W_in[u], w1 = W_in[HID + u], w2 = W_in[2 * HID + u], w3 = W_in[3 * HID + u];
        float z   = xn[0] * w0 + xn[1] * w1 + xn[2] * w2 + xn[3] * w3 + b_in[u];
        float zt0 = s4[0] * w0, zt1 = s4[1] * w1, zt2 = s4[2] * w2, zt3 = s4[3] * w3;
        float h = tanhf(z);
        float g = 1.0f - h * h;
        float ht0 = g * zt0, ht1 = g * zt1, ht2 = g * zt2, ht3 = g * zt3;
        acc[t][0] = h;
        acc[t][1] = ht0; acc[t][2] = ht1; acc[t][3] = ht2; acc[t][4] = ht3;
        acc[t][5] = -2.0f * h * ht0 * zt0;   // z'' = 0 at the input layer
        acc[t][6] = -2.0f * h * ht1 * zt1;
        acc[t][7] = -2.0f * h * ht2 * zt2;
    }

    float* act = actL + wv * (16 * ASTRIDE);
    const int stBase = ph * 8 * ASTRIDE + l15; // + r*ASTRIDE + 16*t
    const int aBase  = l15 * ASTRIDE + kd;     // + k_global

    // ---- 7 hidden layers: jets batched through 256x256 GEMM on f32 WMMA ----
    for (int l = 0; l < NHIDL; ++l) {
        // publish this wave's activations into A-operand layout (row=(p,ch), u-major)
#pragma unroll
        for (int t = 0; t < 16; ++t) {
#pragma unroll
            for (int r = 0; r < 8; ++r) {
                act[stBase + r * ASTRIDE + t * 16] = acc[t][r];
            }
        }
        float bias[16];
#pragma unroll
        for (int t = 0; t < 16; ++t) bias[t] = b_hid[l * HID + t * 16 + l15];

        const v8f zero8 = {0.f, 0.f, 0.f, 0.f, 0.f, 0.f, 0.f, 0.f};
#pragma unroll
        for (int t = 0; t < 16; ++t) acc[t] = zero8;

        const float* Wl = W_hid + l * HID * HID;
        for (int s = 0; s < HID / KSLAB; ++s) {
            const int buf = s & 1;
            // slab s fully arrived in this wave's view...
            asm volatile("s_wait_asynccnt 0x0" ::: "memory");
            // ...and in everyone's; also: everyone finished reading buf[(s+1)&1]
            __syncthreads();
            // issue next async batch (runs concurrently with the WMMAs below)
            if (s < HID / KSLAB - 1)      stage(Wl, s + 1, (s + 1) & 1);
            else if (l + 1 < NHIDL)       stage(Wl + HID * HID, 0, 0);

            const float* wb = smem + ACT_DW + buf * WL_DW;
            // K-loop over the slab: one A chunk feeds 16 column-tile WMMAs
            for (int k0 = 0; k0 < KSLAB; k0 += 4) {
                v2f a = *(const v2f*)(act + aBase + s * KSLAB + k0);
#pragma unroll
                for (int t = 0; t < 16; ++t) {
                    v2f b = *(const v2f*)(wb + (t * 16 + l15) * WSTRIDE + k0 + kd);
                    acc[t] = __builtin_amdgcn_wmma_f32_16x16x4_f32(
                        false, a, false, b, (short)0, acc[t], false, false);
                }
            }
        }
        // tanh-jet epilogue: all 8 channels of (point,unit) live in this lane
        // (overlaps the already-issued async stage of the next layer's slab 0)
#pragma unroll
        for (int t = 0; t < 16; ++t) {
            float z    = acc[t][0] + bias[t];
            float zt0  = acc[t][1], zt1 = acc[t][2], zt2 = acc[t][3], zt3 = acc[t][4];
            float ztt0 = acc[t][5], ztt1 = acc[t][6], ztt2 = acc[t][7];
            float h = tanhf(z);
            float g = 1.0f - h * h;
            float ht0 = g * zt0, ht1 = g * zt1, ht2 = g * zt2, ht3 = g * zt3;
            acc[t][0] = h;
            acc[t][1] = ht0; acc[t][2] = ht1; acc[t][3] = ht2; acc[t][4] = ht3;
            acc[t][5] = g * ztt0 - 2.0f * h * ht0 * zt0;
            acc[t][6] = g * ztt1 - 2.0f * h * ht1 * zt1;
            acc[t][7] = g * ztt2 - 2.0f * h * ht2 * zt2;
        }
    }

    // publish final hidden-layer jets for the output stage
#pragma unroll
    for (int t = 0; t < 16; ++t) {
#pragma unroll
        for (int r = 0; r < 8; ++r) {
            act[stBase + r * ASTRIDE + t * 16] = acc[t][r];
        }
    }
    __syncthreads();   // wL0 free for reuse; act visible

    // ---- output layer (256 -> 4) on all 8 jet channels ----
    const int ch = (lane >> 2) & 7;
    const int j  = lane & 3;
    const float* row0 = actL + wv * (16 * ASTRIDE) + ch * ASTRIDE;  // point 0
    const float* row1 = row0 + 8 * ASTRIDE;                          // point 1
    float o0 = 0.f, o1 = 0.f;
    for (int u = 0; u < HID; ++u) {
        float wo = W_out[u * 4 + j];
        o0 = fmaf(row0[u], wo, o0);
        o1 = fmaf(row1[u], wo, o1);
    }
    float* outB = wL0;                   // [wave][p][ch][j] -> 4*2*8*4 floats
    outB[wv * 64 +      ch * 4 + j] = o0;
    outB[wv * 64 + 32 + ch * 4 + j] = o1;

    float* resB = wL0 + 256;             // [8 points][2]
    if (l15 == 0) {                      // lanes 0 and 16 finish their point
        float J[8][4];
#pragma unroll
        for (int c = 0; c < 8; ++c)
#pragma unroll
            for (int q = 0; q < 4; ++q)
                J[c][q] = outB[wv * 64 + ph * 32 + c * 4 + q];
        float uu  = J[0][0] + b_out[0];
        float vv  = J[0][1] + b_out[1];
        float wwv = J[0][2] + b_out[2];
        float invRe = 1.0f / fmaxf(Re[0], 1e-8f);
        float f[3];
#pragma unroll
        for (int i = 0; i < 3; ++i) {
            float lap = J[5][i] + J[6][i] + J[7][i];
            // jac[i,3] + u*jac[i,0] + v*jac[i,1] + w*jac[i,2] + jac[3,i] - lap/Re
            f[i] = J[4][i] + uu * J[1][i] + vv * J[2][i] + wwv * J[3][i]
                 + J[1 + i][3] - invRe * lap;
        }
        float fe  = J[1][0] + J[2][1] + J[3][2];
        float spP = sqrtf(uu * uu + vv * vv + wwv * wwv) * 0.125f;
        float a0 = uvw[myp * 3 + 0], a1 = uvw[myp * 3 + 1], a2 = uvw[myp * 3 + 2];
        float sp = sqrtf(a0 * a0 + a1 * a1 + a2 * a2) * 0.125f;
        float d  = sp - spP;
        resB[(wv * 2 + ph) * 2 + 0] = d * d;
        resB[(wv * 2 + ph) * 2 + 1] = f[0] * f[0] + f[1] * f[1] + f[2] * f[2] + fe * fe;
    }
    __syncthreads();
    if (tid == 0) {   // deterministic per-block reduction
        float sd = 0.f, se = 0.f;
        for (int i = 0; i < PTS_PER_BLK; ++i) { sd += resB[i * 2]; se += resB[i * 2 + 1]; }
        partials[blockIdx.x * 2 + 0] = sd;
        partials[blockIdx.x * 2 + 1] = se;
    }
}

// --------------------------------------------------------------- finalize ---
__global__ void __launch_bounds__(256) finalize_kernel(const float* __restrict__ partials,
                                                       float* __restrict__ out) {
    __shared__ float sd[256], se[256];
    const int tid = threadIdx.x;
    float d = 0.f, e = 0.f;
    for (int i = tid; i < NBLK; i += 256) { d += partials[i * 2]; e += partials[i * 2 + 1]; }
    sd[tid] = d; se[tid] = e;
    __syncthreads();
    for (int off = 128; off > 0; off >>= 1) {
        if (tid < off) { sd[tid] += sd[tid + off]; se[tid] += se[tid + off]; }
        __syncthreads();
    }
    if (tid == 0) out[0] = 0.5f * (sd[0] + se[0]) / (float)NPTS;
}

// ----------------------------------------------------------------- launch ---
extern "C" void kernel_launch(void* const* d_in, const int* in_sizes, int n_in,
                              void* d_out, int out_size, void* d_ws, size_t ws_size,
                              hipStream_t stream) {
    (void)in_sizes; (void)n_in; (void)out_size; (void)ws_size;
    const float* X     = (const float*)d_in[0];
    const float* uvw   = (const float*)d_in[1];
    const float* W_in  = (const float*)d_in[2];
    const float* b_in  = (const float*)d_in[3];
    const float* W_hid = (const float*)d_in[4];
    const float* b_hid = (const float*)d_in[5];
    const float* W_out = (const float*)d_in[6];
    const float* b_out = (const float*)d_in[7];
    const float* Re    = (const float*)d_in[8];

    float* ws       = (float*)d_ws;
    float* mm       = ws;          // [0..3]=lb, [4..7]=ub
    float* partials = ws + 16;     // 1024 blocks x 2 floats
    float* out      = (float*)d_out;

    minmax_kernel<<<1, 256, 0, stream>>>(X, ws);
    pinn_kernel<<<NBLK, TPB, SMEM_BYTES, stream>>>(X, uvw, W_in, b_in, W_hid, b_hid,
                                                   W_out, b_out, Re, mm, partials);
    finalize_kernel<<<1, 256, 0, stream>>>(partials, out);
}